// NH3D_61177514164979
// MI455X (gfx1250) — compile-verified
//
#include <hip/hip_runtime.h>
#include <stdint.h>

#define TILE 256

// ---------------------------------------------------------------------------
// CDNA5 async global<->LDS helpers (gfx1250; tracked by ASYNCcnt)
// ---------------------------------------------------------------------------
__device__ __forceinline__ void wait_async0() {
#if defined(__has_builtin) && __has_builtin(__builtin_amdgcn_s_wait_asynccnt)
    __builtin_amdgcn_s_wait_asynccnt(0);
#else
    asm volatile("s_wait_asynccnt 0x0" ::: "memory");
#endif
}

// Low 32 bits of a generic pointer into __shared__ are the LDS byte offset
// (flat->LDS mapping takes the low 32 bits, relative to the wave's LDS base).
__device__ __forceinline__ uint32_t lds_off(const void* p) {
    return (uint32_t)(uintptr_t)p;
}

__device__ __forceinline__ void async_load_b128(uint32_t lds_addr, uint32_t voff,
                                                const void* sbase) {
    asm volatile("global_load_async_to_lds_b128 %0, %1, %2 offset:0"
                 :: "v"(lds_addr), "v"(voff), "s"(sbase) : "memory");
}
__device__ __forceinline__ void async_store_b128(uint32_t voff, uint32_t lds_addr,
                                                 void* sbase) {
    asm volatile("global_store_async_from_lds_b128 %0, %1, %2 offset:0"
                 :: "v"(voff), "v"(lds_addr), "s"(sbase) : "memory");
}

// ---------------------------------------------------------------------------
// Neo-Hookean point kernel: closed-form spatial tangent in terms of B = F F^T
// ---------------------------------------------------------------------------
__global__ __launch_bounds__(TILE) void nh3d_kernel(
    const float* __restrict__ F_in, const float* __restrict__ mat_par,
    float* __restrict__ psi_out, float* __restrict__ cau_out,
    float* __restrict__ dd_out, int n_full_tiles, int npts)
{
    __shared__ __align__(16) float sF[TILE * 9];    //  9216 B
    __shared__ __align__(16) float sC[TILE * 6];    //  6144 B
    __shared__ __align__(16) float sD[TILE * 36];   // 36864 B

    const int  t    = threadIdx.x;
    const int  tile = blockIdx.x;
    const long p0   = (long)tile * TILE;
    const bool fast = (tile < n_full_tiles);

    const float c1 = mat_par[0];
    const float c2 = mat_par[1];

    float f[9];

    if (fast) {
        // -- cooperative async copy of TILE*9 floats (9216 B = 576 x b128) --
        const float* gsrc = F_in + p0 * 9;
        const uint32_t fb = lds_off(&sF[0]);
        async_load_b128(fb + (uint32_t)t * 16u,          (uint32_t)t * 16u,          gsrc);
        async_load_b128(fb + (uint32_t)(t + 256) * 16u,  (uint32_t)(t + 256) * 16u,  gsrc);
        if (t < 64)
            async_load_b128(fb + (uint32_t)(t + 512) * 16u, (uint32_t)(t + 512) * 16u, gsrc);
        wait_async0();
        __syncthreads();
#pragma unroll
        for (int k = 0; k < 9; ++k) f[k] = sF[t * 9 + k];  // stride 9: conflict-free
    } else {
        const long p = p0 + t;
        if (p < npts) {
#pragma unroll
            for (int k = 0; k < 9; ++k) f[k] = F_in[p * 9 + k];
        } else {
            f[0] = 1.f; f[1] = 0.f; f[2] = 0.f;
            f[3] = 0.f; f[4] = 1.f; f[5] = 0.f;
            f[6] = 0.f; f[7] = 0.f; f[8] = 1.f;
        }
    }

    // ---- B = F F^T (row-major F[i][j] = f[3i+j]) ----
    const float B00 = f[0]*f[0] + f[1]*f[1] + f[2]*f[2];
    const float B11 = f[3]*f[3] + f[4]*f[4] + f[5]*f[5];
    const float B22 = f[6]*f[6] + f[7]*f[7] + f[8]*f[8];
    const float B01 = f[0]*f[3] + f[1]*f[4] + f[2]*f[5];
    const float B02 = f[0]*f[6] + f[1]*f[7] + f[2]*f[8];
    const float B12 = f[3]*f[6] + f[4]*f[7] + f[5]*f[8];
    const float Bm[3][3] = {{B00, B01, B02}, {B01, B11, B12}, {B02, B12, B22}};

    const float J = f[0]*(f[4]*f[8] - f[5]*f[7])
                  - f[1]*(f[3]*f[8] - f[5]*f[6])
                  + f[2]*(f[3]*f[7] - f[4]*f[6]);
    const float I1   = B00 + B11 + B22;
    const float Jm23 = exp2f((-2.0f / 3.0f) * log2f(J));
    const float invJ = 1.0f / J;

    // scalar coefficients (all F^{-T} contractions cancelled analytically)
    const float c_id  = 2.0f * c1 * Jm23;
    const float c_fg  = -(2.0f / 3.0f) * c_id;
    const float c_gg  = (2.0f / 9.0f) * c_id * I1 + 2.0f * c2 * J * (2.0f * J - 1.0f);
    const float c_tau = -(1.0f / 3.0f) * c_id * I1 + 2.0f * c2 * J * (J - 1.0f);
    const float psi   = c1 * (Jm23 * I1 - 3.0f) + c2 * (J - 1.0f) * (J - 1.0f);

    const int II[6] = {0, 1, 2, 0, 0, 1};
    const int JJ[6] = {0, 1, 2, 1, 2, 2};

    float cau[6];
#pragma unroll
    for (int a = 0; a < 6; ++a) {
        const int i = II[a], j = JJ[a];
        cau[a] = (c_id * Bm[i][j] + ((i == j) ? c_tau : 0.0f)) * invJ;
    }

    __align__(16) float dd[36];
#pragma unroll
    for (int a = 0; a < 6; ++a) {
        const int i = II[a], j = JJ[a];
#pragma unroll
        for (int b = 0; b < 6; ++b) {
            const int k = II[b], l = JJ[b];
            float v = 0.0f;
            if (i == j && k == l) v += c_gg;
            if (k == l)           v += c_fg * Bm[i][j];
            if (i == j)           v += c_fg * Bm[k][l];
            float g = 0.0f;
            if (j == l) g += Bm[i][k];
            if (j == k) g += Bm[i][l];
            if (i == l) g += Bm[j][k];
            if (i == k) g += Bm[j][l];
            v += 0.5f * c_id * g;
            dd[a * 6 + b] = v * invJ;
        }
    }

    if (fast) {
        psi_out[p0 + t] = psi;  // stride-1, already coalesced

        // stage C and D point-major (== global linear order)
#pragma unroll
        for (int a = 0; a < 6; ++a) sC[t * 6 + a] = cau[a];
#pragma unroll
        for (int i = 0; i < 9; ++i)  // 16B-aligned b128 LDS writes
            *(float4*)&sD[t * 36 + 4 * i] = *(const float4*)&dd[4 * i];
        __syncthreads();

        // coalesced async drain: LDS linear -> global linear
        float* gc = cau_out + p0 * 6;   // 6144 B = 384 x b128
        float* gd = dd_out + p0 * 36;   // 36864 B = 2304 x b128
        const uint32_t cb = lds_off(&sC[0]);
        const uint32_t db = lds_off(&sD[0]);

        async_store_b128((uint32_t)t * 16u, cb + (uint32_t)t * 16u, gc);
        if (t < 128)
            async_store_b128((uint32_t)(t + 256) * 16u, cb + (uint32_t)(t + 256) * 16u, gc);
#pragma unroll
        for (int i = 0; i < 9; ++i) {
            const uint32_t off = (uint32_t)(t + i * 256) * 16u;
            async_store_b128(off, db + off, gd);
        }
        wait_async0();  // (S_ENDPGM also waits idle; explicit for clarity)
    } else {
        const long p = p0 + t;
        if (p < npts) {
            psi_out[p] = psi;
#pragma unroll
            for (int a = 0; a < 6; ++a) cau_out[p * 6 + a] = cau[a];
#pragma unroll
            for (int a = 0; a < 36; ++a) dd_out[p * 36 + a] = dd[a];
        }
    }
}

// ---------------------------------------------------------------------------
extern "C" void kernel_launch(void* const* d_in, const int* in_sizes, int n_in,
                              void* d_out, int out_size, void* d_ws, size_t ws_size,
                              hipStream_t stream) {
    const float* F_in = (const float*)d_in[0];
    const float* mat  = (const float*)d_in[1];

    const int npts  = in_sizes[0] / 9;
    const int full  = npts / TILE;
    const int tiles = (npts + TILE - 1) / TILE;

    float* out = (float*)d_out;
    float* psi = out;                          // npts
    float* cau = out + (size_t)npts;           // 6*npts
    float* dd  = out + (size_t)npts * 7;       // 36*npts

    nh3d_kernel<<<tiles, TILE, 0, stream>>>(F_in, mat, psi, cau, dd, full, npts);
}